// EmbeddingBagCollection_15676630630539
// MI455X (gfx1250) — compile-verified
//
#include <hip/hip_runtime.h>

// EmbeddingBagCollection pooled-sum for MI455X (gfx1250, wave32).
// One wave per (feature, bag); lane l owns bytes [8l, 8l+8) of the 256B row.
// Rows are staged in LDS via CDNA5 async global->LDS copies with a 4-deep
// ring (prefetch distance 3), synchronized with s_wait_asynccnt (ASYNCcnt,
// in-order completion). Index stream is wave-uniform -> scalar s_load path.

namespace {
constexpr int kF = 8;        // tables
constexpr int kB = 8192;     // bags per feature (2^13)
constexpr int kN = 200000;   // rows per table
constexpr int kD = 64;       // floats per row (256 bytes)
constexpr int kT = 163840;   // jagged indices per feature
constexpr int kWavesPerBlock = 8;
constexpr int kThreads = kWavesPerBlock * 32;        // 256
constexpr int kBlocks = (kF * kB) / kWavesPerBlock;  // 8192 blocks, 65536 waves
constexpr int kBufs = 4;     // LDS row buffers per wave (ring)
constexpr int kDist = 3;     // prefetch distance (rows in flight beyond current)
}

#define ASPACE_GLOBAL __attribute__((address_space(1)))
#define ASPACE_LDS    __attribute__((address_space(3)))

// Exact pointer types expected by the b64 async builtin: pointer to 2 x i32.
typedef int ebc_v2i __attribute__((ext_vector_type(2)));
typedef ebc_v2i ASPACE_GLOBAL* ebc_g2ptr;
typedef ebc_v2i ASPACE_LDS*    ebc_l2ptr;

#if defined(__has_builtin)
#  if __has_builtin(__builtin_amdgcn_global_load_async_to_lds_b64)
#    define EBC_HAS_ASYNC 1
#  else
#    define EBC_HAS_ASYNC 0
#  endif
#  if __has_builtin(__builtin_amdgcn_s_wait_asynccnt)
#    define EBC_HAS_WAITASYNC 1
#  else
#    define EBC_HAS_WAITASYNC 0
#  endif
#else
#  define EBC_HAS_ASYNC 0
#  define EBC_HAS_WAITASYNC 0
#endif

// 8 bytes per lane, 32 lanes => one 256B embedding row per wave per issue.
__device__ __forceinline__ void ebc_async_row_b64(const float* gsrc, float* ldst) {
#if EBC_HAS_ASYNC
  __builtin_amdgcn_global_load_async_to_lds_b64(
      (ebc_g2ptr)(unsigned long long)gsrc,
      // flat addr low 32 bits == LDS byte offset (ISA 10.2 aperture rule)
      (ebc_l2ptr)(unsigned)(unsigned long long)ldst,
      /*imm offset=*/0, /*cpol=*/0);
#else
  unsigned lds_off = (unsigned)(unsigned long long)ldst;
  asm volatile("global_load_async_to_lds_b64 %0, %1, off"
               :: "v"(lds_off), "v"(gsrc) : "memory");
#endif
}

template <int N>
__device__ __forceinline__ void ebc_wait_async_le() {
#if EBC_HAS_WAITASYNC
  __builtin_amdgcn_s_wait_asynccnt(N);
#else
  asm volatile("s_wait_asynccnt %0" :: "i"(N) : "memory");
#endif
  asm volatile("" ::: "memory");  // keep the ds_load below the wait
}

__global__ __launch_bounds__(kThreads) void ebc_pool_kernel(
    const float* __restrict__ tables,   // [F, N, D]
    const int*   __restrict__ values,   // [F, T]
    const int*   __restrict__ offsets,  // [F, B+1]
    float*       __restrict__ out)      // [B, F*D]
{
  // kBufs row buffers per wave: 8 waves * 4 * 256B = 8KB LDS.
  __shared__ __align__(16) float lds_rows[kWavesPerBlock * kBufs * kD];

  const int lane = (int)(threadIdx.x & 31u);
  const int wave = __builtin_amdgcn_readfirstlane((int)(threadIdx.x >> 5));
  const int w    = (int)blockIdx.x * kWavesPerBlock + wave;   // 0..65535
  const int f    = w >> 13;          // w / kB
  const int b    = w & (kB - 1);     // w % kB

  const int* offs  = offsets + f * (kB + 1) + b;
  const int  start = __builtin_amdgcn_readfirstlane(offs[0]);
  const int  end   = __builtin_amdgcn_readfirstlane(offs[1]);

  const int*   vals = values + f * kT;
  const float* tab  = tables + (size_t)f * ((size_t)kN * kD);

  float*    buf = lds_rows + wave * (kBufs * kD);
  const int lo2 = lane * 2;           // this lane's float offset in a row

  float ax = 0.0f, ay = 0.0f;

  // Prologue: put rows start .. min(start+kDist-1, end-1) in flight.
#pragma unroll
  for (int k = 0; k < kDist; ++k) {
    const int r = start + k;
    if (r < end) {
      const int idx = __builtin_amdgcn_readfirstlane(vals[r]);
      ebc_async_row_b64(tab + (size_t)idx * kD + lo2,
                        buf + (r & (kBufs - 1)) * kD + lo2);
    }
  }

  // Invariant at top of iteration t: rows t .. min(t+kDist-1, end-1) in flight.
  for (int t = start; t < end; ++t) {
    const int np = t + kDist;
    if (np < end) {
      const int idx = __builtin_amdgcn_readfirstlane(vals[np]);
      ebc_async_row_b64(tab + (size_t)idx * kD + lo2,
                        buf + (np & (kBufs - 1)) * kD + lo2);
      ebc_wait_async_le<kDist>();      // kDist+1 in flight -> oldest (t) landed
    } else {
      const int rem = end - 1 - t;     // rows in flight beyond row t: 0..kDist-1
      if (rem == 2)      ebc_wait_async_le<2>();
      else if (rem == 1) ebc_wait_async_le<1>();
      else               ebc_wait_async_le<0>();
    }
    const float2 v = *(const float2*)(buf + (t & (kBufs - 1)) * kD + lo2);
    ax += v.x;
    ay += v.y;
  }

  // out[b, f*64 + 2*lane .. +1]; empty bags store zeros (matches segment_sum).
  float2* dst = (float2*)(out + (size_t)b * (kF * kD) + (size_t)f * kD) + lane;
  *dst = make_float2(ax, ay);
}

extern "C" void kernel_launch(void* const* d_in, const int* in_sizes, int n_in,
                              void* d_out, int out_size, void* d_ws, size_t ws_size,
                              hipStream_t stream) {
  (void)in_sizes; (void)n_in; (void)out_size; (void)d_ws; (void)ws_size;
  const float* tables  = (const float*)d_in[0];  // [F,N,D] f32
  const int*   values  = (const int*)d_in[1];    // [F,T] i32
  const int*   offsets = (const int*)d_in[2];    // [F,B+1] i32
  float*       out     = (float*)d_out;          // [B, F*D] f32

  ebc_pool_kernel<<<dim3(kBlocks), dim3(kThreads), 0, stream>>>(
      tables, values, offsets, out);
}